// MHA_74156905333403
// MI455X (gfx1250) — compile-verified
//
#include <hip/hip_runtime.h>
#include <math.h>

#define SEQ 512
#define DM  512
#define DK  64
#define QK_SCALE 0.125f   // 1/sqrt(64)

typedef __attribute__((ext_vector_type(2))) float v2f;
typedef __attribute__((ext_vector_type(8))) float v8f;

// ---------- fp32 WMMA 16x16 tile helpers (V_WMMA_F32_16X16X4_F32) ----------
// A fragment 16x4 (MxK): lanes 0-15 -> M=lane, VGPR0/1 = K0/K1;
//                        lanes 16-31 -> M=lane-16, VGPR0/1 = K2/K3.
__device__ inline v2f load_a_frag(const float* A, int lda, int row0, int k0, int lane) {
  int m  = lane & 15;
  int kk = k0 + ((lane >> 4) << 1);
  const float* p = A + (size_t)(row0 + m) * lda + kk;
  v2f a; a.x = p[0]; a.y = p[1];
  return a;
}

// B fragment 4x16 (KxN), B row-major [K x N]: mirrored layout of A.
__device__ inline v2f load_b_frag(const float* B, int ldb, int k0, int col0, int lane) {
  int n  = lane & 15;
  int kk = k0 + ((lane >> 4) << 1);
  v2f b;
  b.x = B[(size_t)(kk    ) * ldb + col0 + n];
  b.y = B[(size_t)(kk + 1) * ldb + col0 + n];
  return b;
}

// B = M^T where M is [N x K] row-major (used for Q·K^T: M = K-matrix).
__device__ inline v2f load_bt_frag(const float* M, int ldm, int k0, int col0, int lane) {
  int n  = lane & 15;
  int kk = k0 + ((lane >> 4) << 1);
  const float* p = M + (size_t)(col0 + n) * ldm + kk;
  v2f b; b.x = p[0]; b.y = p[1];
  return b;
}

__device__ inline v8f wmma_f32(v2f a, v2f b, v8f c) {
  // 8 args: (neg_a, A, neg_b, B, c_mod, C, reuse_a, reuse_b)
  return __builtin_amdgcn_wmma_f32_16x16x4_f32(false, a, false, b, (short)0, c,
                                               false, false);
}

// C/D layout: VGPR r -> lanes 0-15: (M=r, N=lane); lanes 16-31: (M=r+8, N=lane-16)
__device__ inline void store_c(float* C, int ldc, int row0, int col0, int lane, v8f c) {
  int n  = lane & 15;
  int mo = (lane >> 4) * 8;
#pragma unroll
  for (int r = 0; r < 8; ++r)
    C[(size_t)(row0 + r + mo) * ldc + col0 + n] = c[r];
}

// ---------------- Kernel 1: Q/K/V projections + W_O row-block reduction ----
__global__ __launch_bounds__(512)
void mha_proj_kernel(const float* __restrict__ xq, const float* __restrict__ xk,
                     const float* __restrict__ WQ, const float* __restrict__ WK,
                     const float* __restrict__ WV, const float* __restrict__ WO,
                     float* __restrict__ Q, float* __restrict__ K,
                     float* __restrict__ V, float* __restrict__ WOe) {
  const int wave = threadIdx.x >> 5;
  const int lane = threadIdx.x & 31;
  const int row0 = blockIdx.x * 16;

  if (wave < 12) {
    const int which = wave >> 2;              // 0:Q 1:K 2:V
    const int col0  = (wave & 3) * 16;
    const float* X = (which == 0) ? xq : xk;
    const float* W = (which == 0) ? WQ : (which == 1) ? WK : WV;
    float*       O = (which == 0) ? Q  : (which == 1) ? K  : V;
    v8f acc = {};
#pragma unroll 4
    for (int k = 0; k < DM; k += 4) {
      v2f a = load_a_frag(X, DM, row0, k, lane);
      v2f b = load_b_frag(W, DK, k, col0, lane);
      acc = wmma_f32(a, b, acc);
    }
    store_c(O, DK, row0, col0, lane, acc);
  } else {
    // W_O_eff[j][n] = sum_{h=0..7} W_O[h*64 + j][n]   (tiled-heads collapse)
    const int tl = threadIdx.x - 384;          // 0..127
    const int base = blockIdx.x * 1024;        // 32 blocks * 1024 = 64*512
    for (int i = tl; i < 1024; i += 128) {
      const int e = base + i;
      const int j = e >> 9;                    // 0..63
      const int n = e & 511;
      float s = 0.f;
#pragma unroll
      for (int h = 0; h < 8; ++h)
        s += WO[(size_t)(h * 64 + j) * DM + n];
      WOe[e] = s;
    }
  }
}

// ---------------- Kernel 2: fused scores -> softmax -> P·V -> out ----------
__global__ __launch_bounds__(512)
void mha_attn_kernel(const float* __restrict__ Q, const float* __restrict__ Km,
                     const float* __restrict__ V, const float* __restrict__ WOe,
                     const float* __restrict__ mask, float* __restrict__ out) {
  __shared__ float S[16][SEQ + 4];   // scores / probabilities for 16 query rows
  __shared__ float Hd[16][DK + 8];   // head tile

  const int wave = threadIdx.x >> 5;
  const int lane = threadIdx.x & 31;
  const int q0   = blockIdx.x * 16;
  const float NEG_INF = -__builtin_inff();

  // ---- Stage 1: scores[16][512] = SCALE * Q_tile (16x64) @ K^T (64x512)
#pragma unroll
  for (int t = 0; t < 2; ++t) {
    const int col0 = (wave * 2 + t) * 16;
    v8f acc = {};
#pragma unroll
    for (int k = 0; k < DK; k += 4) {
      v2f a = load_a_frag(Q, DK, q0, k, lane);
      v2f b = load_bt_frag(Km, DK, k, col0, lane);
      acc = wmma_f32(a, b, acc);
    }
    const int n  = lane & 15;
    const int mo = (lane >> 4) * 8;
#pragma unroll
    for (int r = 0; r < 8; ++r)
      S[r + mo][col0 + n] = acc[r] * QK_SCALE;
  }
  __syncthreads();

  // ---- Stage 2: masked softmax, one wave per query row
  {
    const int r = wave;
    float vals[16];
    float vmax = NEG_INF;
#pragma unroll
    for (int i = 0; i < 16; ++i) {
      const int c = lane + i * 32;
      float v = S[r][c];
      if (mask[c] == 0.0f) v = NEG_INF;
      vals[i] = v;
      vmax = fmaxf(vmax, v);
    }
#pragma unroll
    for (int off = 16; off >= 1; off >>= 1)
      vmax = fmaxf(vmax, __shfl_xor(vmax, off, 32));
    float sum = 0.f;
#pragma unroll
    for (int i = 0; i < 16; ++i) {
      vals[i] = __expf(vals[i] - vmax);
      sum += vals[i];
    }
#pragma unroll
    for (int off = 16; off >= 1; off >>= 1)
      sum += __shfl_xor(sum, off, 32);
    const float inv = 1.0f / sum;
#pragma unroll
    for (int i = 0; i < 16; ++i)
      S[r][lane + i * 32] = vals[i] * inv;
  }
  __syncthreads();

  // ---- Stage 3: head[16][64] = P (16x512, LDS) @ V (512x64)
  if (wave < 4) {                         // wave-uniform branch: EXEC stays all-1s
    const int col0 = wave * 16;
    v8f acc = {};
    for (int k = 0; k < SEQ; k += 4) {
      v2f a = load_a_frag(&S[0][0], SEQ + 4, 0, k, lane);
      v2f b = load_b_frag(V, DK, k, col0, lane);
      acc = wmma_f32(a, b, acc);
    }
    const int n  = lane & 15;
    const int mo = (lane >> 4) * 8;
#pragma unroll
    for (int r = 0; r < 8; ++r)
      Hd[r + mo][col0 + n] = acc[r];
  }
  __syncthreads();

  // ---- Stage 4: out[16][512] = head (16x64, LDS) @ W_O_eff (64x512)
#pragma unroll
  for (int t = 0; t < 2; ++t) {
    const int col0 = (wave * 2 + t) * 16;
    v8f acc = {};
#pragma unroll
    for (int k = 0; k < DK; k += 4) {
      v2f a = load_a_frag(&Hd[0][0], DK + 8, 0, k, lane);
      v2f b = load_b_frag(WOe, DM, k, col0, lane);
      acc = wmma_f32(a, b, acc);
    }
    store_c(out, DM, q0, col0, lane, acc);
  }
}

// ---------------------------------------------------------------------------
extern "C" void kernel_launch(void* const* d_in, const int* in_sizes, int n_in,
                              void* d_out, int out_size, void* d_ws, size_t ws_size,
                              hipStream_t stream) {
  const float* xq   = (const float*)d_in[0];
  const float* xk   = (const float*)d_in[1];
  const float* mask = (const float*)d_in[2];
  const float* WQ   = (const float*)d_in[3];
  const float* WK   = (const float*)d_in[4];
  const float* WV   = (const float*)d_in[5];
  const float* WO   = (const float*)d_in[6];

  float* ws  = (float*)d_ws;
  float* Q   = ws;                       // [512,64]
  float* K   = ws + (size_t)SEQ * DK;    // [512,64]
  float* V   = ws + (size_t)2 * SEQ * DK;// [512,64]
  float* WOe = ws + (size_t)3 * SEQ * DK;// [64,512]

  mha_proj_kernel<<<32, 512, 0, stream>>>(xq, xk, WQ, WK, WV, WO, Q, K, V, WOe);
  mha_attn_kernel<<<32, 512, 0, stream>>>(Q, K, V, WOe, mask, (float*)d_out);
}